// ChamferDistanceL2_38268158607577
// MI455X (gfx1250) — compile-verified
//
#include <hip/hip_runtime.h>
#include <hip/hip_bf16.h>

typedef __attribute__((ext_vector_type(2))) float v2f;
typedef __attribute__((ext_vector_type(8))) float v8f;

#define BATCH 8
#define NPTS 8192
#define ROWS_PER_BLOCK 128   // 8 waves * 16 rows
#define FLT_BIG 3.0e38f

// Pack a point cloud: q[i] = (-2x, -2y, -2z, x^2+y^2+z^2).
// Serves as both the A-form (x = -0.5*q.xyz, sq = q.w) and the B-form
// (-2y vector + 1, with sq used as the C accumulator init).
__global__ void chamfer_pack_kernel(const float* __restrict__ xyz,
                                    float4* __restrict__ q, int total) {
    int i = blockIdx.x * blockDim.x + threadIdx.x;
    if (i < total) {
        float x = xyz[3 * i + 0];
        float y = xyz[3 * i + 1];
        float z = xyz[3 * i + 2];
        q[i] = make_float4(-2.0f * x, -2.0f * y, -2.0f * z, x * x + y * y + z * z);
    }
}

__global__ void chamfer_zero_kernel(float* a) {
    if (threadIdx.x < 2) a[threadIdx.x] = 0.0f;
}

// For each row point p in P, min over all column points q in Q of
// ||p - q||^2, summed over rows, atomically accumulated into *accum.
// One wave = 16 rows; one V_WMMA_F32_16X16X4_F32 = one 16x16 distance tile.
__global__ void __launch_bounds__(256)
chamfer_min_kernel(const float4* __restrict__ P, const float4* __restrict__ Q,
                   float* __restrict__ accum, int np, int nq) {
    const int lane  = threadIdx.x & 31;
    const int wave  = threadIdx.x >> 5;
    const int b     = blockIdx.y;
    const int row0  = blockIdx.x * ROWS_PER_BLOCK + wave * 16;
    const int col   = lane & 15;   // row index for A, column index for B/C/D
    const int khalf = lane >> 4;   // 0: K=0,1   1: K=2,3

    // ---- A fragment (16x4 f32, 2 VGPRs), loop-invariant ----
    // row i of A = (x, y, z, ||x||^2); packed form is (-2x,-2y,-2z,||x||^2)
    float4 p = P[(size_t)b * np + row0 + col];
    v2f a;
    a.x = (khalf == 0) ? (-0.5f * p.x) : (-0.5f * p.z);
    a.y = (khalf == 0) ? (-0.5f * p.y) : p.w;

    float rmin[8];
#pragma unroll
    for (int r = 0; r < 8; ++r) rmin[r] = FLT_BIG;

    const float4* Qb = Q + (size_t)b * nq;

#pragma unroll 2
    for (int t = 0; t < nq; t += 16) {
        float4 q = Qb[t + col];
        // prefetch a future column chunk (global_prefetch_b8)
        if (t + 512 < nq) __builtin_prefetch(Qb + t + 512 + col, 0, 1);

        // ---- B fragment (4x16 f32, 2 VGPRs) ----
        // col j of B = (-2y0, -2y1, -2y2, 1)
        v2f bf;
        bf.x = (khalf == 0) ? q.x : q.z;
        bf.y = (khalf == 0) ? q.y : 1.0f;

        // ---- C init: broadcast ||y_j||^2 down each column ----
        float s = q.w;
        v8f c = {s, s, s, s, s, s, s, s};

        // D[i,j] = ||x_i||^2 - 2 x_i . y_j + ||y_j||^2
        v8f d = __builtin_amdgcn_wmma_f32_16x16x4_f32(
            false, a, false, bf, (short)0, c, false, false);

#pragma unroll
        for (int r = 0; r < 8; ++r) rmin[r] = fminf(rmin[r], d[r]);
    }

    // Cross-lane min within each 16-lane half (rows r / r+8), then sum rows.
    float sum = 0.0f;
#pragma unroll
    for (int r = 0; r < 8; ++r) {
        float v = rmin[r];
        v = fminf(v, __shfl_xor(v, 1, 32));
        v = fminf(v, __shfl_xor(v, 2, 32));
        v = fminf(v, __shfl_xor(v, 4, 32));
        v = fminf(v, __shfl_xor(v, 8, 32));
        sum += v;
    }
    // lanes 0-15 hold sum of rows 0..7 mins, lanes 16-31 rows 8..15
    sum += __shfl_xor(sum, 16, 32);

    if (lane == 0) atomicAdd(accum, sum);
}

__global__ void chamfer_finalize_kernel(const float* __restrict__ acc,
                                        float* __restrict__ out) {
    if (threadIdx.x == 0) {
        const float inv1 = 1.0f / ((float)BATCH * (float)NPTS);
        const float inv2 = 1.0f / ((float)BATCH * (float)NPTS);
        out[0] = acc[0] * inv1 + acc[1] * inv2;
    }
}

extern "C" void kernel_launch(void* const* d_in, const int* in_sizes, int n_in,
                              void* d_out, int out_size, void* d_ws, size_t ws_size,
                              hipStream_t stream) {
    const float* xyz1 = (const float*)d_in[0];
    const float* xyz2 = (const float*)d_in[1];
    float* out = (float*)d_out;

    const int total = BATCH * NPTS;
    float4* q1 = (float4*)d_ws;          // BATCH*NPTS packed points of cloud 1
    float4* q2 = q1 + total;             // BATCH*NPTS packed points of cloud 2
    float* accum = (float*)(q2 + total); // [0]=sum dist1, [1]=sum dist2

    chamfer_zero_kernel<<<1, 32, 0, stream>>>(accum);

    const int pb = 256;
    chamfer_pack_kernel<<<(total + pb - 1) / pb, pb, 0, stream>>>(xyz1, q1, total);
    chamfer_pack_kernel<<<(total + pb - 1) / pb, pb, 0, stream>>>(xyz2, q2, total);

    dim3 grid(NPTS / ROWS_PER_BLOCK, BATCH);
    // dist1: rows from cloud1, min over cloud2
    chamfer_min_kernel<<<grid, 256, 0, stream>>>(q1, q2, accum + 0, NPTS, NPTS);
    // dist2: rows from cloud2, min over cloud1
    chamfer_min_kernel<<<grid, 256, 0, stream>>>(q2, q1, accum + 1, NPTS, NPTS);

    chamfer_finalize_kernel<<<1, 32, 0, stream>>>(accum, out);
}